// YOLOLoss_44787918963006
// MI455X (gfx1250) — compile-verified
//
#include <hip/hip_runtime.h>
#include <math.h>

// ---------------------------------------------------------------------------
// YOLO loss for MI455X (gfx1250, wave32).
// Memory-bound streaming reduction: 328MB read once -> ~14us at 23.3 TB/s.
// WMMA f32 16x16x4 used as the cross-lane reducer (2 scalars per 2 wmma ops).
// ---------------------------------------------------------------------------

typedef __attribute__((ext_vector_type(2))) float v2f;
typedef __attribute__((ext_vector_type(8))) float v8f;

#if defined(__has_builtin)
#  if __has_builtin(__builtin_amdgcn_wmma_f32_16x16x4_f32)
#    define HAVE_WMMA_F32_X4 1
#  endif
#endif

// Reduce two scalars (d, c) across the 32 lanes of a wave.
// Returns a value such that: lane 0 holds sum_lanes(d), lane 16 holds sum_lanes(c).
//
// WMMA1: A (16x4): vgpr0=d, vgpr1=c  => A[m][0]=d[m], A[m][1]=c[m],
//        A[m][2]=d[m+16], A[m][3]=c[m+16]   (per ISA 32-bit A layout)
//        B (4x16): columns n<8 select even K (d), n>=8 select odd K (c)
//        => D1[m][n<8] = d[m]+d[m+16], D1[m][n>=8] = c[m]+c[m+16]
// VALU:  s0 = D1 vgpr0..3 summed, s1 = vgpr4..7 summed (partial row groups)
// WMMA2: A=(s0,s1), B=ones => D2[m][*] = total_d (m<8 rows) / total_c (m>=8),
//        so D2 vgpr0: lanes 0-15 = total_d, lanes 16-31 = total_c.
__device__ __forceinline__ float wave_reduce_pair(float d, float c) {
    const int lane = threadIdx.x & 31;
#if defined(HAVE_WMMA_F32_X4)
    const float sel = ((lane & 15) < 8) ? 1.0f : 0.0f;
    v2f A;    A[0] = d;          A[1] = c;
    v2f Bsel; Bsel[0] = sel;     Bsel[1] = 1.0f - sel;
    v8f Z = {};
    v8f D1 = __builtin_amdgcn_wmma_f32_16x16x4_f32(
        false, A, false, Bsel, (short)0, Z, false, false);
    v2f A2;   A2[0] = D1[0] + D1[1] + D1[2] + D1[3];
              A2[1] = D1[4] + D1[5] + D1[6] + D1[7];
    v2f Bone; Bone[0] = 1.0f;    Bone[1] = 1.0f;
    v8f D2 = __builtin_amdgcn_wmma_f32_16x16x4_f32(
        false, A2, false, Bone, (short)0, Z, false, false);
    return D2[0];
#else
    // Fallback: plain butterfly (should not be needed on gfx1250).
    #pragma unroll
    for (int off = 16; off > 0; off >>= 1) {
        d += __shfl_xor(d, off, 32);
        c += __shfl_xor(c, off, 32);
    }
    return (lane < 16) ? d : c;
#endif
}

__device__ __forceinline__ float fsigmoid(float x) {
    return 1.0f / (1.0f + __expf(-x));
}

#define GRID   20
#define NCELL  400
#define NA     5

// ---------------------------------------------------------------------------
// Kernel 1: stream all B*400 cells. Thread -> 4 consecutive cells (float4
// loads per channel => perfectly coalesced 512B/wave/channel). Accumulate:
//   d  = sum over 4 cells of sum_k (decoded_k - ag_k)^2   (k = x,y,w,h)
//   c2 = sum over 4 cells of conf^2
// Per-block partials (2 floats) written to ws.
// ---------------------------------------------------------------------------
__global__ __launch_bounds__(256) void k_cells(const float* __restrict__ pred,
                                               const float* __restrict__ anchors,
                                               float* __restrict__ partial,
                                               int total /* = B*100 */) {
    const int t = blockIdx.x * 256 + threadIdx.x;   // one group of 4 cells
    float d = 0.0f, c2 = 0.0f;
    if (t < total) {
        const int b = t / 100;            // image
        const int g = t % 100;            // cell-group (4 cells)
        const float* base = pred + (size_t)b * (25 * NCELL) + g * 4;

        float acc[5][4];
        #pragma unroll
        for (int j = 0; j < 5; ++j)
            #pragma unroll
            for (int k = 0; k < 4; ++k) acc[j][k] = 0.0f;

        #pragma unroll
        for (int a = 0; a < NA; ++a) {
            #pragma unroll
            for (int j = 0; j < 5; ++j) {
                const float4 v =
                    *reinterpret_cast<const float4*>(base + (a * 5 + j) * NCELL);
                acc[j][0] += v.x; acc[j][1] += v.y;
                acc[j][2] += v.z; acc[j][3] += v.w;
            }
        }

        const float aw = anchors[0], ah = anchors[1];
        const float agxy = 0.5f / (float)GRID;   // 0.025
        #pragma unroll
        for (int k = 0; k < 4; ++k) {
            const float p0 = acc[0][k] * 0.2f, p1 = acc[1][k] * 0.2f;
            const float p2 = acc[2][k] * 0.2f, p3 = acc[3][k] * 0.2f;
            const float p4 = acc[4][k] * 0.2f;
            const float x = fsigmoid(p0) * (1.0f / (float)GRID);
            const float y = fsigmoid(p1) * (1.0f / (float)GRID);
            const float w = __expf(p2) * aw;
            const float h = __expf(p3) * ah;
            const float cf = fsigmoid(p4);
            const float dx = x - agxy, dy = y - agxy, dw = w - aw, dh = h - ah;
            d  += dx * dx + dy * dy + dw * dw + dh * dh;
            c2 += cf * cf;
        }
    }

    // wave reduce (WMMA) + cross-wave LDS reduce
    const float r = wave_reduce_pair(d, c2);
    __shared__ float sred[16];                 // 8 waves * {d, c2}
    const int lane = threadIdx.x & 31, wv = threadIdx.x >> 5;
    if (lane == 0)  sred[wv * 2 + 0] = r;
    if (lane == 16) sred[wv * 2 + 1] = r;
    __syncthreads();
    if (threadIdx.x == 0) {
        float sd = 0.0f, sc = 0.0f;
        #pragma unroll
        for (int w = 0; w < 8; ++w) { sd += sred[w * 2]; sc += sred[w * 2 + 1]; }
        partial[blockIdx.x * 2 + 0] = sd;
        partial[blockIdx.x * 2 + 1] = sc;
    }
}

// ---------------------------------------------------------------------------
// Kernel 2: per-image obj-cell path. Thread b: gather 25 floats at the obj
// cell, decode, IoU vs label. Accumulate 4 scalars:
//   S1 = (px-lx)^2+(py-ly)^2 + (sqrt(pw)-sqrt(lw))^2+(sqrt(ph)-sqrt(lh))^2
//   S2 = d_obj  (same metric as kernel 1's d, at the obj cell)
//   S3 = (conf - IoU)^2
//   S4 = conf^2
// ---------------------------------------------------------------------------
__global__ __launch_bounds__(256) void k_obj(const float* __restrict__ pred,
                                             const float* __restrict__ label,
                                             const float* __restrict__ anchors,
                                             float* __restrict__ partial,
                                             int B) {
    const int b = blockIdx.x * 256 + threadIdx.x;
    float S1 = 0.0f, S2 = 0.0f, S3 = 0.0f, S4 = 0.0f;
    if (b < B) {
        const float lx = label[b * 4 + 0], ly = label[b * 4 + 1];
        const float lw = label[b * 4 + 2], lh = label[b * 4 + 3];
        const int gx = (int)floorf(lx * (float)GRID);
        const int gy = (int)floorf(ly * (float)GRID);
        const int idx = gx * GRID + gy;

        const float* base = pred + (size_t)b * (25 * NCELL) + idx;
        float acc[5] = {0.f, 0.f, 0.f, 0.f, 0.f};
        #pragma unroll
        for (int a = 0; a < NA; ++a)
            #pragma unroll
            for (int j = 0; j < 5; ++j)
                acc[j] += base[(a * 5 + j) * NCELL];

        const float aw = anchors[0], ah = anchors[1];
        const float agxy = 0.5f / (float)GRID;
        const float px = fsigmoid(acc[0] * 0.2f) * (1.0f / (float)GRID);
        const float py = fsigmoid(acc[1] * 0.2f) * (1.0f / (float)GRID);
        const float pw = __expf(acc[2] * 0.2f) * aw;
        const float ph = __expf(acc[3] * 0.2f) * ah;
        const float pc = fsigmoid(acc[4] * 0.2f);

        // coordinate (obj) loss pieces
        const float dx = px - lx, dy = py - ly;
        const float dsw = __fsqrt_rn(pw) - __fsqrt_rn(lw);
        const float dsh = __fsqrt_rn(ph) - __fsqrt_rn(lh);
        S1 = dx * dx + dy * dy + dsw * dsw + dsh * dsh;

        // d_obj (to subtract from d_all)
        const float ex = px - agxy, ey = py - agxy, ew = pw - aw, eh = ph - ah;
        S2 = ex * ex + ey * ey + ew * ew + eh * eh;

        // IoU
        const float lx1 = fmaxf(lx - lw * 0.5f, 0.f), ly1 = fmaxf(ly - lh * 0.5f, 0.f);
        const float lx2 = fminf(lx + lw * 0.5f, 1.f), ly2 = fminf(ly + lh * 0.5f, 1.f);
        const float px1 = fmaxf(px - pw * 0.5f, 0.f), py1 = fmaxf(py - ph * 0.5f, 0.f);
        const float px2 = fminf(px + pw * 0.5f, 1.f), py2 = fminf(py + ph * 0.5f, 1.f);
        const float iw = fmaxf(fminf(lx2, px2) - fmaxf(lx1, px1), 0.f);
        const float ih = fmaxf(fminf(ly2, py2) - fmaxf(ly1, py1), 0.f);
        const float inter = iw * ih;
        const float iou = inter / (lw * lh + pw * ph - inter);

        const float dc = pc - iou;
        S3 = dc * dc;
        S4 = pc * pc;
    }

    const float r12 = wave_reduce_pair(S1, S2);
    const float r34 = wave_reduce_pair(S3, S4);
    __shared__ float sred[32];                 // 8 waves * {S1,S2,S3,S4}
    const int lane = threadIdx.x & 31, wv = threadIdx.x >> 5;
    if (lane == 0)  { sred[wv * 4 + 0] = r12; sred[wv * 4 + 2] = r34; }
    if (lane == 16) { sred[wv * 4 + 1] = r12; sred[wv * 4 + 3] = r34; }
    __syncthreads();
    if (threadIdx.x == 0) {
        float a0 = 0.f, a1 = 0.f, a2 = 0.f, a3 = 0.f;
        #pragma unroll
        for (int w = 0; w < 8; ++w) {
            a0 += sred[w * 4 + 0]; a1 += sred[w * 4 + 1];
            a2 += sred[w * 4 + 2]; a3 += sred[w * 4 + 3];
        }
        partial[blockIdx.x * 4 + 0] = a0;
        partial[blockIdx.x * 4 + 1] = a1;
        partial[blockIdx.x * 4 + 2] = a2;
        partial[blockIdx.x * 4 + 3] = a3;
    }
}

// ---------------------------------------------------------------------------
// Kernel 3: combine partials, apply loss weights, write scalar.
// ---------------------------------------------------------------------------
__global__ __launch_bounds__(256) void k_final(const float* __restrict__ p1,
                                               const float* __restrict__ p2,
                                               float* __restrict__ out,
                                               int nblk1, int nblk2, int B) {
    __shared__ float sd[256], sc[256];
    float dall = 0.0f, call = 0.0f;
    for (int i = threadIdx.x; i < nblk1; i += 256) {
        dall += p1[2 * i + 0];
        call += p1[2 * i + 1];
    }
    sd[threadIdx.x] = dall;
    sc[threadIdx.x] = call;
    __syncthreads();
    for (int s = 128; s > 0; s >>= 1) {
        if (threadIdx.x < s) {
            sd[threadIdx.x] += sd[threadIdx.x + s];
            sc[threadIdx.x] += sc[threadIdx.x + s];
        }
        __syncthreads();
    }
    if (threadIdx.x == 0) {
        float S1 = 0.f, S2 = 0.f, S3 = 0.f, S4 = 0.f;
        for (int i = 0; i < nblk2; ++i) {
            S1 += p2[4 * i + 0]; S2 += p2[4 * i + 1];
            S3 += p2[4 * i + 2]; S4 += p2[4 * i + 3];
        }
        const float d_all = sd[0], c_all = sc[0];
        const float fB = (float)B;
        const float ncm1 = (float)(NCELL - 1);
        const float coor_obj   = S1 / (2.0f * fB);
        const float coor_noobj = (d_all - S2) / (fB * ncm1 * 4.0f);
        const float conf_obj   = S3 / fB;
        const float conf_noobj = (c_all - S4) / (fB * ncm1);
        out[0] = 1.5f * coor_obj + 1.0f * coor_noobj
               + 5.0f * conf_obj + 1.0f * conf_noobj;
    }
}

extern "C" void kernel_launch(void* const* d_in, const int* in_sizes, int n_in,
                              void* d_out, int out_size, void* d_ws, size_t ws_size,
                              hipStream_t stream) {
    const float* pred    = (const float*)d_in[0];   // [B, 25, 20, 20] f32
    const float* label   = (const float*)d_in[1];   // [B, 4] f32
    const float* anchors = (const float*)d_in[2];   // [5, 2] f32
    float* out = (float*)d_out;

    const int B      = in_sizes[0] / (25 * NCELL);  // 8192
    const int total1 = B * (NCELL / 4);             // 4-cell groups
    const int nblk1  = (total1 + 255) / 256;        // 3200
    const int nblk2  = (B + 255) / 256;             // 32

    float* partial1 = (float*)d_ws;                 // nblk1 * 2 floats
    float* partial2 = partial1 + (size_t)nblk1 * 2; // nblk2 * 4 floats

    k_cells<<<nblk1, 256, 0, stream>>>(pred, anchors, partial1, total1);
    k_obj  <<<nblk2, 256, 0, stream>>>(pred, label, anchors, partial2, B);
    k_final<<<1, 256, 0, stream>>>(partial1, partial2, out, nblk1, nblk2, B);
}